// Attention_747324309931
// MI455X (gfx1250) — compile-verified
//
#include <hip/hip_runtime.h>
#include <hip/hip_bf16.h>

// Problem sizes (fixed by the reference)
constexpr int B = 128;
constexpr int L = 1024;
constexpr int E = 512;
constexpr int D = 512;
constexpr int A = 512;

typedef __attribute__((ext_vector_type(2))) float v2f;
typedef __attribute__((ext_vector_type(8))) float v8f;
typedef __attribute__((ext_vector_type(4))) unsigned int u32x4;
typedef __attribute__((ext_vector_type(8))) int i32x8;
typedef __attribute__((ext_vector_type(4))) int i32x4;

// ---------------------------------------------------------------------------
// Kernel 1: s2[b,a] = sum_d decoder_hidden[b,d] * W_dec[d,a] + b_enc[a] + b_dec[a]
// ---------------------------------------------------------------------------
__global__ __launch_bounds__(256) void att2_kernel(
    const float* __restrict__ dh,    // [B, D]
    const float* __restrict__ Wdec,  // [D, A]
    const float* __restrict__ benc,  // [A]
    const float* __restrict__ bdec,  // [A]
    float* __restrict__ s2)          // [B, A]
{
    const int idx = blockIdx.x * 256 + threadIdx.x;   // B*A = 65536
    const int b = idx >> 9;
    const int a = idx & (A - 1);
    const float* dhb = dh + b * D;
    float acc = 0.f;
    #pragma unroll 4
    for (int d = 0; d < D; ++d)
        acc = fmaf(dhb[d], Wdec[d * A + a], acc);
    s2[idx] = acc + benc[a] + bdec[a];
}

// ---------------------------------------------------------------------------
// Kernel 2 (main, WMMA + TDM):
//   scores[b,l] = relu(enc[b,l,:]@W_enc + s2[b,:]) . W_full + b_full
//   grid  : B * (L/32) = 4096 blocks; block = 256 threads = 8 wave32s.
//   Each wave owns a 64-col strip (4 N-tiles) and TWO 16-row M-tiles that
//   share the same B fragments (halves W_enc traffic per WMMA).
//   The 32x512 fp32 encoder tile (contiguous 64 KB) is DMA'd into LDS by the
//   Tensor Data Mover; the LDS region is reused for the row reduction after.
// ---------------------------------------------------------------------------
__global__ __launch_bounds__(256) void scores_kernel(
    const float* __restrict__ enc,    // [B, L, E]
    const float* __restrict__ Wenc,   // [E, A]
    const float* __restrict__ s2,     // [B, A]
    const float* __restrict__ Wfull,  // [A]
    const float* __restrict__ bfull,  // [1]
    float* __restrict__ scores)       // [B, L]
{
    __shared__ __align__(16) float smem[32 * 512];   // 64 KB tile, reused for reduction

    const int tid = threadIdx.x;
    const int b   = blockIdx.x >> 5;         // 32 row-tiles of 32 rows per batch
    const int l0  = (blockIdx.x & 31) << 5;

    const float* gsrc = enc + ((size_t)b * L + l0) * E;   // contiguous 32*512 floats

#if __has_builtin(__builtin_amdgcn_tensor_load_to_lds)
    if (tid < 32) {  // wave 0 issues one TDM descriptor for the whole 64 KB tile
        const unsigned long long gaddr = (unsigned long long)(size_t)gsrc;
        const unsigned int      laddr  = (unsigned int)(size_t)smem;
        const unsigned int      td0    = (unsigned int)B * L * E;   // 1-D tensor length (elems)
        const unsigned int      tile0  = 32u * 512u;                // tile length (elems)

        u32x4 g0 = {};
        g0[0] = 1u;                                            // count=1, user descriptor
        g0[1] = laddr;                                         // lds_addr (bytes)
        g0[2] = (unsigned int)gaddr;                           // global_addr[31:0]
        g0[3] = (unsigned int)((gaddr >> 32) & 0x01FFFFFFull)  // global_addr[56:32]
              | (2u << 30);                                    // type = 2 ("image")

        i32x8 g1 = {};
        g1[0] = (int)(2u << 16);                               // data_size = 4 bytes
        g1[1] = (int)((td0 & 0xFFFFu) << 16);                  // tensor_dim0[15:0]
        g1[2] = (int)(((td0 >> 16) & 0xFFFFu) | (1u << 16));   // tensor_dim0[31:16], tensor_dim1=1
        g1[3] = (int)(tile0 << 16);                            // tile_dim0 = 16384
        g1[4] = 0;                                             // tile_dim1/2 unused
        g1[5] = (int)td0;                                      // tensor_dim0_stride[31:0]
        g1[6] = 0;
        g1[7] = 0;

        i32x4 g2 = {};
        i32x4 g3 = {};
        i32x8 g4 = {};   // 6-arg toolchain form: extra descriptor group (unused)
        __builtin_amdgcn_tensor_load_to_lds(g0, g1, g2, g3, g4, 0);
        __builtin_amdgcn_s_wait_tensorcnt(0);
    }
#else
    {
        const float4* src4 = (const float4*)gsrc;
        float4* dst4 = (float4*)smem;
        #pragma unroll
        for (int it = 0; it < 16; ++it)
            dst4[tid + it * 256] = src4[tid + it * 256];
    }
#endif
    __syncthreads();

    const int wid   = tid >> 5;
    const int lane  = tid & 31;
    const int l16   = lane & 15;
    const int lhalf = lane >> 4;             // 0: K+{0,1}, 1: K+{2,3}
    const int colbase = wid * 64;            // this wave's 64-col strip

    // A-fragment sources in LDS for the two M-tiles (rows l16 and l16+16).
    const float* ldsA0 = &smem[l16 * 512 + 2 * lhalf];
    const float* ldsA1 = &smem[(l16 + 16) * 512 + 2 * lhalf];
    // B-fragment source: W_enc[k0 + 2*lhalf + {0,1}][colbase + t*16 + l16]
    const float* Bp = Wenc + (size_t)(2 * lhalf) * A + colbase + l16;

    v8f acc[2][4] = {};

    for (int k0 = 0; k0 < E; k0 += 4) {
        v2f a0, a1;
        a0.x = ldsA0[k0]; a0.y = ldsA0[k0 + 1];
        a1.x = ldsA1[k0]; a1.y = ldsA1[k0 + 1];
        const float* bp = Bp + (size_t)k0 * A;
        v2f bv[4];
        #pragma unroll
        for (int t = 0; t < 4; ++t) {
            bv[t].x = bp[16 * t];
            bv[t].y = bp[A + 16 * t];
        }
        #pragma unroll
        for (int t = 0; t < 4; ++t) {
            acc[0][t] = __builtin_amdgcn_wmma_f32_16x16x4_f32(
                false, a0, false, bv[t], (short)0, acc[0][t], false, false);
            acc[1][t] = __builtin_amdgcn_wmma_f32_16x16x4_f32(
                false, a1, false, bv[t], (short)0, acc[1][t], false, false);
        }
    }

    // Fused epilogue: v = relu(att1 + s2) * W_full, accumulated per row.
    // C layout: VGPR i, lanes 0-15 -> row i, lanes 16-31 -> row i+8; col = t*16 + l16.
    const float* s2b = s2 + b * A + colbase + l16;
    const float* wfb = Wfull + colbase + l16;
    float part[2][8];
    #pragma unroll
    for (int mt = 0; mt < 2; ++mt)
        #pragma unroll
        for (int i = 0; i < 8; ++i) part[mt][i] = 0.f;

    #pragma unroll
    for (int t = 0; t < 4; ++t) {
        const float sv = s2b[t * 16];
        const float wv = wfb[t * 16];
        #pragma unroll
        for (int mt = 0; mt < 2; ++mt) {
            #pragma unroll
            for (int i = 0; i < 8; ++i) {
                float v = acc[mt][t][i] + sv;
                v = v > 0.f ? v : 0.f;
                part[mt][i] = fmaf(v, wv, part[mt][i]);
            }
        }
    }

    // Tile is dead: reuse LDS for the cross-wave row reduction (8x32x16 floats).
    __syncthreads();
    float (*red)[32][16] = (float (*)[32][16])smem;
    #pragma unroll
    for (int mt = 0; mt < 2; ++mt)
        #pragma unroll
        for (int i = 0; i < 8; ++i)
            red[wid][lane][mt * 8 + i] = part[mt][i];
    __syncthreads();

    // Row r in [0,32): M-tile mt=r>>4, r16=r&15 -> VGPR i=r16&7, lane half r16>>3.
    if (tid < 32) {
        const int mt  = tid >> 4;
        const int r16 = tid & 15;
        const int i   = mt * 8 + (r16 & 7);
        const int hl  = (r16 >> 3) << 4;
        float s = *bfull;   // softmax-invariant, kept for fidelity
        for (int w = 0; w < 8; ++w) {
            #pragma unroll
            for (int l2 = 0; l2 < 16; ++l2)
                s += red[w][hl + l2][i];
        }
        scores[b * L + l0 + tid] = s;
    }
}

// ---------------------------------------------------------------------------
// Kernel 3: alpha[b,:] = softmax(scores[b,:]) over L=1024. One block per batch.
// ---------------------------------------------------------------------------
__global__ __launch_bounds__(256) void softmax_kernel(
    const float* __restrict__ scores,  // [B, L]
    float* __restrict__ alpha)         // [B, L]
{
    __shared__ float sm[256];
    const int b = blockIdx.x;
    const int tid = threadIdx.x;

    float v[4];
    float m = -INFINITY;
    #pragma unroll
    for (int j = 0; j < 4; ++j) {
        v[j] = scores[b * L + tid + j * 256];
        m = fmaxf(m, v[j]);
    }
    sm[tid] = m;
    __syncthreads();
    for (int s = 128; s > 0; s >>= 1) {
        if (tid < s) sm[tid] = fmaxf(sm[tid], sm[tid + s]);
        __syncthreads();
    }
    const float mx = sm[0];
    __syncthreads();

    float sum = 0.f;
    #pragma unroll
    for (int j = 0; j < 4; ++j) {
        v[j] = __expf(v[j] - mx);
        sum += v[j];
    }
    sm[tid] = sum;
    __syncthreads();
    for (int s = 128; s > 0; s >>= 1) {
        if (tid < s) sm[tid] += sm[tid + s];
        __syncthreads();
    }
    const float inv = 1.0f / sm[0];
    #pragma unroll
    for (int j = 0; j < 4; ++j)
        alpha[b * L + tid + j * 256] = v[j] * inv;
}

// ---------------------------------------------------------------------------
// Kernel 4: weighted[b,e] = sum_l enc[b,l,e] * alpha[b,l].
// ---------------------------------------------------------------------------
__global__ __launch_bounds__(256) void weighted_kernel(
    const float* __restrict__ enc,    // [B, L, E]
    const float* __restrict__ alpha,  // [B, L]
    float* __restrict__ out)          // [B, E]
{
    const int b = blockIdx.x >> 1;
    const int e = ((blockIdx.x & 1) << 8) + threadIdx.x;
    const float* ep = enc + (size_t)b * L * E + e;
    const float* ap = alpha + b * L;
    float acc = 0.f;
    #pragma unroll 8
    for (int l = 0; l < L; ++l)
        acc = fmaf(ep[(size_t)l * E], ap[l], acc);
    out[b * E + e] = acc;
}

// ---------------------------------------------------------------------------
extern "C" void kernel_launch(void* const* d_in, const int* in_sizes, int n_in,
                              void* d_out, int out_size, void* d_ws, size_t ws_size,
                              hipStream_t stream) {
    const float* enc   = (const float*)d_in[0];  // [B,L,E]
    const float* dh    = (const float*)d_in[1];  // [B,D]
    const float* Wenc  = (const float*)d_in[2];  // [E,A]
    const float* benc  = (const float*)d_in[3];  // [A]
    const float* Wdec  = (const float*)d_in[4];  // [D,A]
    const float* bdec  = (const float*)d_in[5];  // [A]
    const float* Wfull = (const float*)d_in[6];  // [A]
    const float* bfull = (const float*)d_in[7];  // [1]

    float* weighted = (float*)d_out;             // [B,E]
    float* alpha    = (float*)d_out + B * E;     // [B,L]

    float* s2     = (float*)d_ws;                // [B,A]   256 KB
    float* scores = s2 + B * A;                  // [B,L]   512 KB

    att2_kernel<<<(B * A) / 256, 256, 0, stream>>>(dh, Wdec, benc, bdec, s2);
    scores_kernel<<<B * (L / 32), 256, 0, stream>>>(enc, Wenc, s2, Wfull, bfull, scores);
    softmax_kernel<<<B, 256, 0, stream>>>(scores, alpha);
    weighted_kernel<<<B * (E / 256), 256, 0, stream>>>(enc, alpha, weighted);
}